// SpikeToCalcium_54030688584266
// MI455X (gfx1250) — compile-verified
//
#include <hip/hip_runtime.h>
#include <hip/hip_bf16.h>

typedef float v2f __attribute__((ext_vector_type(2)));
typedef float v8f __attribute__((ext_vector_type(8)));
typedef int   v4i __attribute__((ext_vector_type(4)));

#define U_W            30000
#define OUT_W          29940
#define KLEN           61
#define NKK            19        // ceil((61+15)/4) K-steps of 4
#define ROWS_PER_BLOCK 16
#define COLS_PER_BLOCK 256       // 8 waves * 2 tiles * 16 cols
#define LOAD_COLS      320       // 256 + 60 halo, rounded up to 64
#define LDS_STRIDE     324       // multiple of 4 (16B-aligned rows for async B128);
                                 // 324 % 64 == 4 -> banks 4m+c distinct for m=0..15,
                                 // lane-half offset +2 has opposite parity -> conflict-free

#if defined(__gfx1250__) && __has_builtin(__builtin_amdgcn_global_load_async_to_lds_b128)
#define USE_ASYNC_LDS 1
#else
#define USE_ASYNC_LDS 0
#endif

__global__ __launch_bounds__(256)
void SpikeToCalcium_conv_wmma(const float* __restrict__ u,
                              const float* __restrict__ kern,
                              float* __restrict__ out, int rows) {
    __shared__ float su[ROWS_PER_BLOCK * LDS_STRIDE];
    __shared__ float wp[96];     // wp[15+t] = kern[KLEN-1-t], zeros elsewhere

    const int tid     = threadIdx.x;
    const int cb      = blockIdx.x * COLS_PER_BLOCK;   // first output col of block
    const int rowBase = blockIdx.y * ROWS_PER_BLOCK;

    // ---- stage reversed, zero-padded weights ----
    if (tid < 96) {
        float w = 0.0f;
        int t = tid - 15;
        if (t >= 0 && t < KLEN) w = kern[KLEN - 1 - t];
        wp[tid] = w;
    }

    // ---- stage input tile: 16 rows x 320 cols ----
    {
        const int r  = tid >> 4;            // 0..15  (one row per 16 threads)
        const int c4 = (tid & 15) * 4;      // 0,4,...,60
        const float* urow = u + (size_t)(rowBase + r) * U_W;
        #pragma unroll
        for (int i = 0; i < 5; ++i) {
            int col  = c4 + i * 64;         // 0..319
            int gcol = cb + col;
            float* dst = &su[r * LDS_STRIDE + col];   // 16B aligned (stride%4==0, col%4==0)
#if USE_ASYNC_LDS
            if (gcol < U_W) {
                // CDNA5 async copy: global -> LDS, no VGPR round-trip (ASYNCcnt)
                __builtin_amdgcn_global_load_async_to_lds_b128(
                    (__attribute__((address_space(1))) v4i*)(float*)(urow + gcol),
                    (__attribute__((address_space(3))) v4i*)dst,
                    0, 0);
            } else {
                dst[0] = 0.f; dst[1] = 0.f; dst[2] = 0.f; dst[3] = 0.f;
            }
#else
            float4 v = make_float4(0.f, 0.f, 0.f, 0.f);
            if (gcol < U_W) v = *(const float4*)(urow + gcol);  // U_W%4==0, cb%4==0
            dst[0] = v.x; dst[1] = v.y; dst[2] = v.z; dst[3] = v.w;
#endif
        }
    }
#if USE_ASYNC_LDS
    // async LDS writes must complete before the barrier publishes them
#if __has_builtin(__builtin_amdgcn_s_wait_asynccnt)
    __builtin_amdgcn_s_wait_asynccnt(0);
#else
    asm volatile("s_wait_asynccnt 0" ::: "memory");
#endif
#endif
    __syncthreads();

    // ---- per-wave: two 16x16 output tiles via V_WMMA_F32_16X16X4_F32 ----
    const int lane  = tid & 31;
    const int wave  = tid >> 5;             // 0..7
    const int n     = lane & 15;            // N (output col within tile) / M (A row)
    const int khalf = (lane >> 4) << 1;     // 0 or 2 : K-split across lane halves

    const float* Arow = &su[n * LDS_STRIDE + wave * 32];  // A row m == n for this lane

    v8f acc0 = {0,0,0,0,0,0,0,0};
    v8f acc1 = {0,0,0,0,0,0,0,0};

    #pragma unroll
    for (int kk = 0; kk < NKK; ++kk) {
        const int j = kk * 4 + khalf;       // A/B K-index handled by this lane half
        v2f a0, a1, b;
        // A fragment: VGPR0 = K=khalf, VGPR1 = K=khalf+1 (ISA 16x4 fp32 layout)
        a0.x = Arow[j];       a0.y = Arow[j + 1];        // tile 0
        a1.x = Arow[j + 16];  a1.y = Arow[j + 17];       // tile 1 (shifted 16 cols)
        // B fragment (banded Toeplitz weights): B[k,n] = wp[15 + k - n]
        const int bi = 15 + j - n;                       // in [0, 90]
        b.x = wp[bi]; b.y = wp[bi + 1];
        acc0 = __builtin_amdgcn_wmma_f32_16x16x4_f32(
                   false, a0, false, b, (short)0, acc0, false, false);
        acc1 = __builtin_amdgcn_wmma_f32_16x16x4_f32(
                   false, a1, false, b, (short)0, acc1, false, false);
    }

    // ---- store: D layout = 8 VGPRs, lanes 0-15 -> M=r, lanes 16-31 -> M=r+8 ----
    const int mAdd  = (lane >> 4) << 3;     // 0 or 8
    const int gcol0 = cb + wave * 32 + n;
    const int gcol1 = gcol0 + 16;
    #pragma unroll
    for (int r = 0; r < 8; ++r) {
        size_t base = (size_t)(rowBase + mAdd + r) * OUT_W;
        if (gcol0 < OUT_W) out[base + gcol0] = acc0[r];
        if (gcol1 < OUT_W) out[base + gcol1] = acc1[r];
    }
}

extern "C" void kernel_launch(void* const* d_in, const int* in_sizes, int n_in,
                              void* d_out, int out_size, void* d_ws, size_t ws_size,
                              hipStream_t stream) {
    const float* u    = (const float*)d_in[0];   // (1024, 30000) fp32
    const float* kern = (const float*)d_in[1];   // (61,) fp32
    float* out = (float*)d_out;                  // (1024, 29940) fp32

    const int rows = in_sizes[0] / U_W;          // 1024
    dim3 grid((OUT_W + COLS_PER_BLOCK - 1) / COLS_PER_BLOCK,   // 117
              rows / ROWS_PER_BLOCK);                          // 64
    SpikeToCalcium_conv_wmma<<<grid, 256, 0, stream>>>(u, kern, out, rows);
}